// WindowAttentionLayer_82437602280000
// MI455X (gfx1250) — compile-verified
//
#include <hip/hip_runtime.h>
#include <hip/hip_bf16.h>

typedef float v2f __attribute__((ext_vector_type(2)));
typedef float v8f __attribute__((ext_vector_type(8)));

#define NB    4
#define NC    128
#define NH_   64
#define NW_   128
#define HIDD  128
#define NHEAD 8
#define DH    16
#define NS    4
#define LTOK  512
#define NWIN  512          // B*NHEAD*NS*NS
#define EPSV  1e-6f
#define KPAD  20           // padded row stride (floats): 16B-aligned + conflict-free

// async copy of one 16-byte chunk global -> LDS (ASYNCcnt-tracked, no VGPR data)
__device__ __forceinline__ void async_b128(const float* g, const float* lds_ptr) {
  unsigned lds = (unsigned)(uintptr_t)lds_ptr;            // low 32 bits = LDS offset
  unsigned long long ga = (unsigned long long)(uintptr_t)g;
  asm volatile("global_load_async_to_lds_b128 %0, %1, off"
               :: "v"(lds), "v"(ga) : "memory");
}
__device__ __forceinline__ void async_wait0() {
  asm volatile("s_wait_asynccnt 0x0" ::: "memory");
}

// ---------------------------------------------------------------------------
// Kernel 1: LayerNorm + 1x1 projections (WMMA f32) + scatter into shifted-
// window layout q/k/v : [win][token][16].  block=256, 32 px/block
// ---------------------------------------------------------------------------
__global__ void k_ln_qkv(const float* __restrict__ f1, const float* __restrict__ f2,
                         const float* __restrict__ g1, const float* __restrict__ b1,
                         const float* __restrict__ g2, const float* __restrict__ b2,
                         const float* __restrict__ Wq, const float* __restrict__ bq,
                         const float* __restrict__ Wkv, const float* __restrict__ bkv,
                         float* __restrict__ qws, float* __restrict__ kws,
                         float* __restrict__ vws, const int* __restrict__ shiftp) {
  __shared__ float buf[32 * 132];          // [px][c] padded
  const int tid  = threadIdx.x;
  const int wid  = tid >> 5;
  const int lane = tid & 31;
  const int hw   = lane >> 4;
  const int ln16 = lane & 15;
  const int pbase = blockIdx.x * 32;
  const int b_ = pbase >> 13;
  const int sp = pbase & 8191;
  const int y  = sp >> 7;
  const int xbase = sp & 127;
  const int shift = *shiftp;
  const int sh = shift ? 8 : 0, sw = shift ? 16 : 0;
  const int koff = hw ? 2 : 0;

  for (int pass = 0; pass < 2; ++pass) {
    const float* f  = pass ? f2 : f1;
    const float* gg = pass ? g2 : g1;
    const float* bb = pass ? b2 : b1;
    __syncthreads();
    for (int i = 0; i < 16; ++i) {
      int idx = tid + i * 256;
      int c = idx >> 5, px = idx & 31;
      buf[px * 132 + c] =
          f[((size_t)(b_ * NC + c) << 13) + (y << 7) + xbase + px];
    }
    __syncthreads();
    {   // LayerNorm, 8 threads per pixel
      int px = tid >> 3, sub = tid & 7;
      float s = 0.f, s2 = 0.f, vals[16];
      for (int k = 0; k < 16; ++k) {
        float v = buf[px * 132 + sub * 16 + k];
        vals[k] = v; s += v; s2 += v * v;
      }
      for (int m = 1; m < 8; m <<= 1) {
        s  += __shfl_xor(s,  m, 32);
        s2 += __shfl_xor(s2, m, 32);
      }
      float mu  = s * (1.f / 128.f);
      float var = s2 * (1.f / 128.f) - mu * mu;
      float rs  = rsqrtf(var + EPSV);
      for (int k = 0; k < 16; ++k) {
        int c = sub * 16 + k;
        buf[px * 132 + c] = (vals[k] - mu) * rs * gg[c] + bb[c];
      }
    }
    __syncthreads();
    const int ntiles = pass ? 32 : 16;
    const float* Wm = pass ? Wkv : Wq;
    const float* bm = pass ? bkv : bq;
    for (int tix = wid; tix < ntiles; tix += 8) {
      int mt = tix & 1, nt = tix >> 1;
      int arow = (mt * 16 + ln16) * 132;
      int brow = (nt * 16 + ln16) * 128;
      // preload full B tile into registers: 32 independent b64 loads
      v2f bw[32];
      {
        const float2* wp2 = (const float2*)Wm;
        int base2 = (brow + koff) >> 1;
        for (int s = 0; s < 32; ++s) {
          float2 t2 = wp2[base2 + (s << 1)];
          bw[s].x = t2.x; bw[s].y = t2.y;
        }
      }
      v8f acc = {};
      for (int s = 0; s < 32; ++s) {
        v2f a;
        a.x = buf[arow + 4 * s + koff];
        a.y = buf[arow + 4 * s + koff + 1];
        acc = __builtin_amdgcn_wmma_f32_16x16x4_f32(false, a, false, bw[s],
                                                    (short)0, acc, false, false);
      }
      int col  = nt * 16 + ln16;
      float bias = bm[col];
      int isv  = (pass && col >= 128);
      int head = (isv ? (col - 128) : col) >> 4;
      int d    = col & 15;
      float* dst = pass ? (isv ? vws : kws) : qws;
      for (int j = 0; j < 8; ++j) {
        int pxl = mt * 16 + j + (hw ? 8 : 0);
        int x = xbase + pxl;
        int yy = (y - sh + NH_) & (NH_ - 1);
        int xx = (x - sw + NW_) & (NW_ - 1);
        int wy = yy >> 4, ly = yy & 15, wx = xx >> 5, lx = xx & 31;
        int t   = ly * 32 + lx;
        int win = ((b_ * NHEAD + head) * NS + wy) * NS + wx;
        dst[((size_t)(win * LTOK + t) << 4) + d] = acc[j] + bias;
      }
    }
  }
}

// ---------------------------------------------------------------------------
// Kernel 2 (fused attention): per 16x16 tile do QK^T WMMA -> scale+mask ->
// store sim (d_out) -> E=exp(S) -> accumulate row sums AND E@V WMMA
// unnormalized (exact: no max-subtraction needed, so no flash rescale).
// Divide by Z once at the end + eps rank-1 correction, scatter into att.
// block=256 (8 waves = 8 row strips), 4 blocks per window.
// ---------------------------------------------------------------------------
__global__ void k_attn(const float* __restrict__ qws, const float* __restrict__ kws,
                       const float* __restrict__ vws, float* __restrict__ sim,
                       float* __restrict__ att, const int* __restrict__ shiftp) {
  __shared__ float lk[LTOK * KPAD];        // 40,960 B (K, padded)
  __shared__ float lv[LTOK * KPAD];        // 40,960 B (V, padded)
  __shared__ float et[8][16 * KPAD];       // 10,240 B (per-wave E bounce tiles)
  __shared__ float csp[16][16];
  __shared__ float cs[16];
  const int tid = threadIdx.x, wid = tid >> 5, lane = tid & 31;
  const int hw = lane >> 4, ln16 = lane & 15;
  const int hw8 = hw ? 8 : 0;
  const int win = blockIdx.x >> 2, mg = blockIdx.x & 3;
  const int shift = *shiftp;
  const int sh = shift ? 8 : 0, sw = shift ? 16 : 0;
  const int koff = hw ? 2 : 0;

  const float* kbase = kws + ((size_t)win << 13);
  const float* vbase = vws + ((size_t)win << 13);
  for (int i = 0; i < 8; ++i) {            // 2x2048 16B chunks / 256 threads
    int cj = tid + i * 256;
    int t = cj >> 2, dp = (cj & 3) << 2;
    async_b128(kbase + (cj << 2), &lk[t * KPAD + dp]);
    async_b128(vbase + (cj << 2), &lv[t * KPAD + dp]);
  }
  async_wait0();
  __syncthreads();
  {  // column sums of V (for the eps rank-1 correction)
    int col = tid & 15, ch = tid >> 4;     // ch in [0,16)
    float s = 0.f;
    for (int r = ch * 32; r < ch * 32 + 32; ++r) s += lv[r * KPAD + col];
    csp[ch][col] = s;
  }
  __syncthreads();
  if (tid < 16) {
    float s = 0.f;
    for (int ch = 0; ch < 16; ++ch) s += csp[ch][tid];
    cs[tid] = s;
  }
  __syncthreads();

  const int mbase = (mg * 8 + wid) * 16;
  // Q strip in A-operand register layout (8 floats/lane)
  const float* qrow = qws + ((size_t)(win * LTOK + mbase + ln16) << 4);
  float aq[8];
  for (int s = 0; s < 4; ++s) {
    aq[2 * s]     = qrow[4 * s + koff];
    aq[2 * s + 1] = qrow[4 * s + koff + 1];
  }

  const int wx = win & 3, wy = (win >> 2) & 3;
  float lsum[8];
  int rlab[8];
  for (int j = 0; j < 8; ++j) {
    lsum[j] = 0.f;
    int ti = mbase + j + hw8;
    int gy = wy * 16 + (ti >> 5), gx = wx * 32 + (ti & 31);
    rlab[j] = ((gy < 48) ? 0 : ((gy < 56) ? 1 : 2)) * 3 +
              ((gx < 96) ? 0 : ((gx < 112) ? 1 : 2));
  }

  float* simw = sim + ((size_t)win << 18);
  float* etw = &et[wid][0];
  v8f acc_o = {};
  for (int nb = 0; nb < LTOK; nb += 16) {
    // ---- S tile = scale*Q@K^T (+mask) ----
    v8f accs = {};
    int bb = (nb + ln16) * KPAD + koff;
    for (int s = 0; s < 4; ++s) {
      v2f a, bv;
      a.x = aq[2 * s];
      a.y = aq[2 * s + 1];
      bv.x = lk[bb + 4 * s];
      bv.y = lk[bb + 4 * s + 1];
      accs = __builtin_amdgcn_wmma_f32_16x16x4_f32(false, a, false, bv,
                                                   (short)0, accs, false, false);
    }
    int tj = nb + ln16;
    int gy = wy * 16 + (tj >> 5), gx = wx * 32 + (tj & 31);
    int clab = ((gy < 48) ? 0 : ((gy < 56) ? 1 : 2)) * 3 +
               ((gx < 96) ? 0 : ((gx < 112) ? 1 : 2));
    // ---- sim store + E = exp(S) into bounce tile + row partial sums ----
    for (int j = 0; j < 8; ++j) {
      float v = accs[j] * 0.25f;
      if (shift && rlab[j] != clab) v += -100.0f;
      simw[((size_t)(mbase + j + hw8) << 9) + tj] = v;
      float e = __expf(v);                 // values O(1): no max-sub needed
      lsum[j] += e;
      etw[(j + hw8) * KPAD + ln16] = e;    // D-layout -> [m][n] tile
    }
    // ---- acc_o += E @ V (unnormalized) ----
    for (int s = 0; s < 4; ++s) {
      v2f a, bv;
      int k0 = 4 * s + koff;
      a.x = etw[ln16 * KPAD + k0];         // A-layout reads of E
      a.y = etw[ln16 * KPAD + k0 + 1];
      int kg = nb + k0;
      bv.x = lv[kg * KPAD + ln16];
      bv.y = lv[(kg + 1) * KPAD + ln16];
      acc_o = __builtin_amdgcn_wmma_f32_16x16x4_f32(false, a, false, bv,
                                                    (short)0, acc_o, false, false);
    }
  }
  // ---- finalize: divide by Z, eps correction, inverse-shift scatter ----
  const float renorm = 1.0f / (1.0f + (float)LTOK * EPSV);
  const int head = (win >> 4) & 7, b_ = win >> 7;
  const float add = EPSV * cs[ln16];
  for (int j = 0; j < 8; ++j) {
    float z = lsum[j];
    for (int m = 1; m < 16; m <<= 1) z += __shfl_xor(z, m, 32);
    float o = acc_o[j] / z + add;          // z replicated across half-wave = row j+hw8
    int t = mbase + j + hw8;
    int gy = wy * 16 + (t >> 5), gx = wx * 32 + (t & 31);
    int oy = (gy + sh) & (NH_ - 1), ox = (gx + sw) & (NW_ - 1);
    att[(((size_t)(b_ * NH_ + oy) * NW_ + ox) << 7) + head * 16 + ln16] =
        o * renorm;
  }
}

// ---------------------------------------------------------------------------
// Kernel 3: out = att @ Wo^T + bo, NCHW stores via LDS micro-transpose.
// block=256 (8 waves = 8 channel tiles), 16 px/block
// ---------------------------------------------------------------------------
__global__ void k_out(const float* __restrict__ att, const float* __restrict__ Wo,
                      const float* __restrict__ bo, float* __restrict__ outp) {
  __shared__ float la[16 * 132];
  __shared__ float lt[8][16 * 17];
  const int tid = threadIdx.x, wid = tid >> 5, lane = tid & 31;
  const int hw = lane >> 4, ln16 = lane & 15;
  const int pbase = blockIdx.x * 16;
  const int koff = hw ? 2 : 0;

  for (int i = tid; i < 2048; i += 256) {
    int px = i >> 7, c = i & 127;
    la[px * 132 + c] = att[(((size_t)pbase + px) << 7) + c];
  }
  __syncthreads();

  const int nt = wid;
  const int brow = (nt * 16 + ln16) * 128;
  const int arow = ln16 * 132;
  v2f bw[32];
  {
    const float2* wp2 = (const float2*)Wo;
    int base2 = (brow + koff) >> 1;
    for (int s = 0; s < 32; ++s) {
      float2 t2 = wp2[base2 + (s << 1)];
      bw[s].x = t2.x; bw[s].y = t2.y;
    }
  }
  v8f acc = {};
  for (int s = 0; s < 32; ++s) {
    v2f a;
    a.x = la[arow + 4 * s + koff];
    a.y = la[arow + 4 * s + koff + 1];
    acc = __builtin_amdgcn_wmma_f32_16x16x4_f32(false, a, false, bw[s],
                                                (short)0, acc, false, false);
  }
  float* t = &lt[wid][0];
  for (int j = 0; j < 8; ++j) {
    int m = j + (hw ? 8 : 0);
    t[ln16 * 17 + m] = acc[j];          // [channel][pixel]
  }
  const int b_ = pbase >> 13, spx = pbase & 8191;
  const int oy = spx >> 7, oxb = spx & 127;
  for (int j2 = 0; j2 < 8; ++j2) {
    int cl = j2 + (hw ? 8 : 0);
    int c  = nt * 16 + cl;
    float v = t[cl * 17 + ln16] + bo[c];
    size_t oidx = (((((size_t)(b_ * NC + c)) << 6) + (size_t)oy) << 7);
    outp[oidx + oxb + ln16] = v;
  }
}

// ---------------------------------------------------------------------------
extern "C" void kernel_launch(void* const* d_in, const int* in_sizes, int n_in,
                              void* d_out, int out_size, void* d_ws, size_t ws_size,
                              hipStream_t stream) {
  const float* f1  = (const float*)d_in[0];
  const float* f2  = (const float*)d_in[1];
  const float* g1  = (const float*)d_in[2];
  const float* b1  = (const float*)d_in[3];
  const float* g2  = (const float*)d_in[4];
  const float* b2  = (const float*)d_in[5];
  const float* Wq  = (const float*)d_in[6];
  const float* bq  = (const float*)d_in[7];
  const float* Wkv = (const float*)d_in[8];
  const float* bkv = (const float*)d_in[9];
  const float* Wo  = (const float*)d_in[10];
  const float* bo  = (const float*)d_in[11];
  const int* shiftp = (const int*)d_in[13];

  float* out = (float*)d_out;
  float* sim = out + (size_t)NB * NC * NH_ * NW_;    // 4,194,304

  const size_t QSZ = (size_t)NWIN * LTOK * DH;       // 4,194,304
  float* w = (float*)d_ws;
  float* qws = w;
  float* kws = qws + QSZ;
  float* vws = kws + QSZ;
  float* attb = vws + QSZ;

  k_ln_qkv<<<1024, 256, 0, stream>>>(f1, f2, g1, b1, g2, b2, Wq, bq, Wkv, bkv,
                                     qws, kws, vws, shiftp);
  k_attn<<<2048, 256, 0, stream>>>(qws, kws, vws, sim, attb, shiftp);
  k_out<<<2048, 256, 0, stream>>>(attb, Wo, bo, out);
}